// RVQECell_74526272520325
// MI455X (gfx1250) — compile-verified
//
#include <hip/hip_runtime.h>
#include <hip/hip_bf16.h>
#include <math.h>

// ---------------------------------------------------------------------------
// RVQE cell, 19 qubits, batch 32, fp32.
//   lane l <-> state bit (18 - l):  inout lanes 0..5 = bits 18..13,
//   workspace lanes 6..17 = bits 12..1, ancilla lane 18 = bit 0.
// Strategy (MI455X, 23.3 TB/s HBM, 192MB L2, wave32):
//   * Precompute 54 per-neuron angle tables (2^17 x {cosA,sinA}) with a
//     WMMA-based z^T M z evaluation (v_wmma_f32_16x16x4_f32).
//   * Fuse each layer into ONE full-state pass over 32KB LDS tiles:
//       pass1: bitflip-gather + 12 input neurons
//       pass2 x3: 12 uniform Ry + 12 kernel neurons (in place)
//       pass3: bitflip-gather + 6 output neurons + probs marginal + psi store
//   * Tiles and table segments move via global_load_async_to_lds_b128 /
//     global_store_async_from_lds_b128 with s_wait_asynccnt gating
//     (double-buffered table prefetch pipeline).
// ---------------------------------------------------------------------------

typedef __attribute__((ext_vector_type(2))) float v2f;
typedef __attribute__((ext_vector_type(8))) float v8f;

#define NQ      19
#define BATCH   32
#define STATE   (1u << 19)
#define TABN    54
#define TABSZ   (1u << 17)
#define BIASPHI 1.57079632679489662f

// low 32 bits of a generic pointer derived from LDS == LDS byte address
__device__ __forceinline__ unsigned ldsAddr(const void* p) {
    return (unsigned)(unsigned long long)p;
}

__device__ __forceinline__ void async_load_b128(unsigned lds, unsigned voff,
                                                unsigned long long sbase) {
    asm volatile("global_load_async_to_lds_b128 %0, %1, %2 offset:0"
                 :: "v"(lds), "v"(voff), "s"(sbase) : "memory");
}
__device__ __forceinline__ void async_store_b128(unsigned voff, unsigned lds,
                                                 unsigned long long sbase) {
    asm volatile("global_store_async_from_lds_b128 %0, %1, %2 offset:0"
                 :: "v"(voff), "v"(lds), "s"(sbase) : "memory");
}
__device__ __forceinline__ void wait_async0() {
    asm volatile("s_wait_asynccnt 0x0" ::: "memory");
}

// ---------------------------------------------------------------------------
// Table build: for neuron n, for every 17-bit control config x:
//   phi = z . (M z),  z = [1, b0..b16], b_i = bit (16-i) of x
//   M[0][0]=bias, M[0][1+i]=th1[i], M[1+i][1+j]=th2[i][j] (j>i)
//   alpha = atan2(sin^2 phi, cos^2 phi):
//     cosA = c^2 / sqrt(c^4+s^4), sinA = s^2 / sqrt(c^4+s^4)
// G = Z @ M^T computed with v_wmma_f32_16x16x4_f32 (16 rows/wave,
// K padded to 20, N padded to 32), then phi = rowdot(Z, G) via shfl_xor.
// ---------------------------------------------------------------------------
__device__ __forceinline__ float zval(int m, int c) {
    if (c == 0) return 1.0f;
    if (c <= 17) return (float)((m >> (17 - c)) & 1);
    return 0.0f;
}

__global__ __launch_bounds__(256)
void build_tables(const float* __restrict__ w_in1, const float* __restrict__ w_in2,
                  const float* __restrict__ w_k1,  const float* __restrict__ w_k2,
                  const float* __restrict__ w_out1,const float* __restrict__ w_out2,
                  float2* __restrict__ tabs) {
    __shared__ float Mlds[32 * 20];  // [n][k], zero padded

    int n     = blockIdx.x >> 10;           // neuron 0..53
    int tbase = (blockIdx.x & 1023) * 128;  // 128 rows per block (8 waves x 16)

    const float* th1; const float* th2;
    if (n < 12)      { th1 = w_in1  + n*17;        th2 = w_in2  + n*289; }
    else if (n < 48) { int k = n-12; th1 = w_k1  + k*17; th2 = w_k2  + k*289; }
    else             { int k = n-48; th1 = w_out1 + k*17; th2 = w_out2 + k*289; }

    for (int e = threadIdx.x; e < 32 * 20; e += 256) {
        int r = e / 20, cc = e % 20;  // r = M row (n-index), cc = M col (k)
        float v = 0.0f;
        if (r == 0)                          v = (cc == 0) ? BIASPHI
                                               : (cc <= 17 ? th1[cc-1] : 0.0f);
        else if (r <= 17 && cc > r && cc <= 17) v = th2[(r-1)*17 + (cc-1)];
        Mlds[e] = v;
    }
    __syncthreads();

    int lane = threadIdx.x & 31;
    int m0   = tbase + (threadIdx.x >> 5) * 16;   // 16 rows per wave

    int mA    = m0 + (lane & 15);
    int kAoff = (lane >> 4) * 2;                  // A: lanes 0-15 K={0,1}, 16-31 K={2,3}
    int nB0   = (lane & 15);                      // B tile0 col
    int nB1   = 16 + (lane & 15);                 // B tile1 col

    v8f acc0 = {};
    v8f acc1 = {};
    for (int s = 0; s < 5; ++s) {                 // K = 20 in steps of 4
        int kb = s * 4 + kAoff;
        v2f a;  a.x  = zval(mA, kb);           a.y  = zval(mA, kb + 1);
        v2f b0; b0.x = Mlds[nB0*20 + kb];      b0.y = Mlds[nB0*20 + kb + 1];
        v2f b1; b1.x = Mlds[nB1*20 + kb];      b1.y = Mlds[nB1*20 + kb + 1];
        acc0 = __builtin_amdgcn_wmma_f32_16x16x4_f32(false, a, false, b0,
                                                     (short)0, acc0, false, false);
        acc1 = __builtin_amdgcn_wmma_f32_16x16x4_f32(false, a, false, b1,
                                                     (short)0, acc1, false, false);
    }

    // epilogue: phi_m = sum_n Z[m,n] * G[m,n]; D layout: lane holds col (lane&15),
    // VGPR r holds row r (lanes 0-15) / 8+r (lanes 16-31).
    int mrow_base = m0 + ((lane >> 4) << 3);
    int ng0 = lane & 15, ng1 = 16 + (lane & 15);
    float phi = 0.0f;
    for (int r = 0; r < 8; ++r) {
        int m = mrow_base + r;
        float t = zval(m, ng0) * acc0[r] + zval(m, ng1) * acc1[r];
        t += __shfl_xor(t, 1);
        t += __shfl_xor(t, 2);
        t += __shfl_xor(t, 4);
        t += __shfl_xor(t, 8);              // all 16 lanes of the half hold row sum
        if ((lane & 7) == r) phi = t;
    }
    // lane owns row m0 + (lane&7) + 8*(lane>=16); lanes with bit3 set are dups
    float sphi = __sinf(phi), cphi = __cosf(phi);
    float s2 = sphi * sphi, c2 = cphi * cphi;
    float rr = rsqrtf(c2 * c2 + s2 * s2);   // >= sqrt(0.5), never 0
    if ((lane & 8) == 0) {
        int mrow = m0 + (lane & 7) + ((lane >> 4) << 3);
        tabs[(size_t)n * TABSZ + mrow] = make_float2(c2 * rr, s2 * rr);
    }
}

// ---------------------------------------------------------------------------
// Fused workspace-layer pass. Tile = fixed (batch b, inout io): 8192 floats,
// contiguous. Optional bitflip-gather on load (input layer), optional uniform
// Ry sweep (kernel stages), then 12 neurons with double-buffered async table
// prefetch. In-place safe: each block touches only its own tile.
// ---------------------------------------------------------------------------
__global__ __launch_bounds__(256)
void work_pass(const float* __restrict__ in, float* __restrict__ out,
               const float2* __restrict__ tabs,   // layer base (12 tables)
               const int* __restrict__ flips,     // null => no bitflip
               const float* __restrict__ uang) {  // null => no unitary sweep
    __shared__ float  tile[8192];
    __shared__ float2 tbuf[2][2048];

    int b  = blockIdx.x >> 6;
    int io = blockIdx.x & 63;
    int tid = threadIdx.x;

    int f = 0;
    if (flips) {
        for (int l = 0; l < 6; ++l) f |= (flips[b * 6 + l] & 1) << (5 - l);
    }

    unsigned long long srcBase =
        (unsigned long long)(in + ((size_t)b << 19) + ((size_t)(io ^ f) << 13));
    unsigned tileL = ldsAddr(tile);
    for (int i = 0; i < 8; ++i) {                     // 8192 floats, B128
        unsigned off = (unsigned)((i * 256 + tid) * 16);
        async_load_b128(tileL + off, off, srcBase);
    }
    // prefetch neuron 0 table segment (2048 float2 = 16KB, contiguous)
    {
        unsigned tb0 = ldsAddr(&tbuf[0][0]);
        unsigned long long sb = (unsigned long long)(tabs + ((size_t)io << 11));
        for (int i = 0; i < 4; ++i) {
            unsigned off = (unsigned)((i * 256 + tid) * 16);
            async_load_b128(tb0 + off, off, sb);
        }
    }
    wait_async0();
    __syncthreads();

    if (uang) {  // UnitaryLayer: Ry(w_u[st][j]) on lane 6+j  <-> state bit 12-j
        for (int j = 0; j < 12; ++j) {
            float cu = __cosf(uang[j]), su = __sinf(uang[j]);
            int bitp = 12 - j;
            for (int i = 0; i < 16; ++i) {
                int p  = i * 256 + tid;
                int lo = p & ((1 << bitp) - 1);
                int t0 = ((p >> bitp) << (bitp + 1)) | lo;
                int t1 = t0 | (1 << bitp);
                float e0 = tile[t0], e1 = tile[t1];
                tile[t0] = cu * e0 - su * e1;
                tile[t1] = su * e0 + cu * e1;
            }
            __syncthreads();
        }
    }

    for (int j = 0; j < 12; ++j) {   // neuron on lane 6+j, pair bit = 12-j
        if (j > 0) { wait_async0(); __syncthreads(); }
        if (j < 11) {                // prefetch next neuron's segment
            unsigned tbn = ldsAddr(&tbuf[(j + 1) & 1][0]);
            unsigned long long sb = (unsigned long long)
                (tabs + (size_t)(j + 1) * TABSZ + ((size_t)io << 11));
            for (int i = 0; i < 4; ++i) {
                unsigned off = (unsigned)((i * 256 + tid) * 16);
                async_load_b128(tbn + off, off, sb);
            }
        }
        const float2* cbuf = tbuf[j & 1];
        int bitp = 12 - j;
        for (int i = 0; i < 16; ++i) {
            int p  = i * 256 + tid;              // pair index, bit0 = ancilla
            int lo = p & ((1 << bitp) - 1);
            int t0 = ((p >> bitp) << (bitp + 1)) | lo;
            int t1 = t0 | (1 << bitp);
            float2 cs = cbuf[p >> 1];            // wsx = removeBit(t,bitp) >> 1
            float e0 = tile[t0], e1 = tile[t1];
            tile[t0] = cs.x * e0 - cs.y * e1;
            tile[t1] = cs.y * e0 + cs.x * e1;
        }
    }
    __syncthreads();

    unsigned long long dstBase =
        (unsigned long long)(out + ((size_t)b << 19) + ((size_t)io << 13));
    for (int i = 0; i < 8; ++i) {
        unsigned off = (unsigned)((i * 256 + tid) * 16);
        async_store_b128(off, tileL + off, dstBase);
    }
    wait_async0();
}

// ---------------------------------------------------------------------------
// Output pass. Tile = fixed (b, chunk c of 128 low bits) x 64 io values.
// Bitflip-gather on load, 6 output neurons (pairs along io bits), probability
// marginal into probs[b][io] via LDS reduce + global atomics, psi store.
// ---------------------------------------------------------------------------
__global__ __launch_bounds__(256)
void out_pass(const float* __restrict__ in, float* __restrict__ psiOut,
              float* __restrict__ probs, const float2* __restrict__ tabs,
              const int* __restrict__ flips) {
    __shared__ float tile[8192];  // [io][t], t in [0,128)
    __shared__ float sp[64];

    int b = blockIdx.x >> 6;
    int c = blockIdx.x & 63;
    int tid = threadIdx.x;

    int f = 0;
    for (int l = 0; l < 6; ++l) f |= (flips[b * 6 + l] & 1) << (5 - l);

    unsigned long long srcBase = (unsigned long long)(in + ((size_t)b << 19));
    unsigned tileL = ldsAddr(tile);
    for (int i = 0; i < 8; ++i) {
        int lin = i * 256 + tid;               // 2048 B128 loads
        int io = lin >> 5, t = (lin & 31) * 4;
        unsigned goff = (unsigned)(((((io ^ f) << 13) | (c * 128 + t))) * 4);
        unsigned loff = (unsigned)((io * 128 + t) * 4);
        async_load_b128(tileL + loff, goff, srcBase);
    }
    if (tid < 64) sp[tid] = 0.0f;
    wait_async0();
    __syncthreads();

    for (int j = 0; j < 6; ++j) {              // neuron on inout lane j, io bit q
        int q = 5 - j;
        for (int i = 0; i < 16; ++i) {
            int p   = i * 256 + tid;           // [iop(5)][t(7)]
            int t   = p & 127;
            int iop = p >> 7;
            int lo  = iop & ((1 << q) - 1);
            int io0 = ((iop >> q) << (q + 1)) | lo;
            int io1 = io0 | (1 << q);
            int ws  = (c * 128 + t) >> 1;      // 12 ws bits (drop ancilla)
            float2 cs = tabs[(size_t)j * TABSZ + ((size_t)iop << 12) + ws];
            float e0 = tile[io0 * 128 + t], e1 = tile[io1 * 128 + t];
            tile[io0 * 128 + t] = cs.x * e0 - cs.y * e1;
            tile[io1 * 128 + t] = cs.y * e0 + cs.x * e1;
        }
        __syncthreads();
    }

    // probs[b][io] += sum_t psi^2
    for (int i = 0; i < 32; ++i) {
        int idx = i * 256 + tid;
        float v = tile[idx];
        atomicAdd(&sp[idx >> 7], v * v);
    }
    __syncthreads();
    if (tid < 64) atomicAdd(&probs[b * 64 + tid], sp[tid]);

    unsigned long long dstBase = (unsigned long long)(psiOut + ((size_t)b << 19));
    for (int i = 0; i < 8; ++i) {
        int lin = i * 256 + tid;
        int io = lin >> 5, t = (lin & 31) * 4;
        unsigned goff = (unsigned)(((io << 13) | (c * 128 + t)) * 4);
        unsigned loff = (unsigned)((io * 128 + t) * 4);
        async_store_b128(goff, tileL + loff, dstBase);
    }
    wait_async0();
}

__global__ void zero_probs(float* p) { p[blockIdx.x * 256 + threadIdx.x] = 0.0f; }

// ---------------------------------------------------------------------------
extern "C" void kernel_launch(void* const* d_in, const int* in_sizes, int n_in,
                              void* d_out, int out_size, void* d_ws, size_t ws_size,
                              hipStream_t stream) {
    const float* batch  = (const float*)d_in[0];
    const float* w_in1  = (const float*)d_in[1];
    const float* w_in2  = (const float*)d_in[2];
    const float* w_u    = (const float*)d_in[3];
    const float* w_k1   = (const float*)d_in[4];
    const float* w_k2   = (const float*)d_in[5];
    const float* w_out1 = (const float*)d_in[6];
    const float* w_out2 = (const float*)d_in[7];
    const int*   inputs = (const int*)d_in[8];

    float* probs = (float*)d_out;              // [32][64]
    float* psi   = probs + BATCH * 64;         // [32][2^19]

    float*  S0   = (float*)d_ws;                                   // 64 MB
    float2* tabs = (float2*)((char*)d_ws + (size_t)BATCH * STATE * 4); // 54 MB

    zero_probs<<<8, 256, 0, stream>>>(probs);
    build_tables<<<dim3(TABN * 1024), 256, 0, stream>>>(
        w_in1, w_in2, w_k1, w_k2, w_out1, w_out2, tabs);

    // input layer: bitflip + 12 neurons (batch -> S0)
    work_pass<<<2048, 256, 0, stream>>>(batch, S0, tabs, inputs, nullptr);
    // 3 kernel stages: unitary + 12 neurons (S0 in place)
    for (int st = 0; st < 3; ++st)
        work_pass<<<2048, 256, 0, stream>>>(
            S0, S0, tabs + (size_t)(12 + st * 12) * TABSZ, nullptr, w_u + st * 12);
    // output: bitflip + 6 neurons + probs + psi
    out_pass<<<2048, 256, 0, stream>>>(
        S0, psi, probs, tabs + (size_t)48 * TABSZ, inputs);

    (void)in_sizes; (void)n_in; (void)out_size; (void)ws_size;
}